// GNN_29291676958840
// MI455X (gfx1250) — compile-verified
//
#include <hip/hip_runtime.h>
#include <hip/hip_bf16.h>

typedef __attribute__((ext_vector_type(2))) float v2f;
typedef __attribute__((ext_vector_type(8))) float v8f;

#define NCOLS 256   // output feature width of every layer

// ---------------- degree / normalization ----------------
__global__ void deg_init_k(float* deg, int n) {
  int i = blockIdx.x * blockDim.x + threadIdx.x;
  if (i < n) deg[i] = 1.0f;                    // self-loop contribution
}

__global__ void deg_count_k(const int* __restrict__ dst, float* deg, int nE) {
  int e = blockIdx.x * blockDim.x + threadIdx.x;
  if (e < nE) atomicAdd(&deg[dst[e]], 1.0f);
}

__global__ void dinv_k(float* deg, int n) {
  int i = blockIdx.x * blockDim.x + threadIdx.x;
  if (i < n) {
    float d = deg[i];
    deg[i] = d > 0.0f ? rsqrtf(d) : 0.0f;      // in place: deg -> dinv
  }
}

// ---------------- H = (relu?)(A) @ W via V_WMMA_F32_16X16X4_F32 ----------------
// A: [M x K] row-major (K = 128 or 256), W: [K x 256], H: [M x 256]
// block = 512 threads = 16 waves; wave w computes the 16x16 tile at columns 16*w.
// A M-tile (16 x K) staged once per block into LDS (row padded +4 to avoid
// 16-way bank conflicts on stride-128/256 column reads).
__global__ __launch_bounds__(512) void gemm_wmma_k(const float* __restrict__ A,
                                                   const float* __restrict__ W,
                                                   float* __restrict__ H,
                                                   int K, int doRelu) {
  __shared__ float As[16 * (256 + 4)];
  const int tid  = threadIdx.x;
  const int lane = tid & 31;
  const int wave = tid >> 5;            // 0..15
  const int m0   = blockIdx.x << 4;     // 16-row tile
  const int col0 = wave << 4;           // 16-col tile per wave
  const int ldsK = K + 4;

  // cooperative LDS stage of the 16 x K A tile, fusing previous layer's relu
  for (int idx = tid; idx < (K << 4); idx += 512) {
    int r = idx / K;
    int c = idx - r * K;
    float v = A[(size_t)(m0 + r) * K + c];
    if (doRelu) v = fmaxf(v, 0.0f);
    As[r * ldsK + c] = v;
  }
  __syncthreads();

  // f32 16x4 A fragment: lanes 0-15 -> K{k,k+1}, lanes 16-31 -> K{k+2,k+3}
  const int r     = lane & 15;          // A row / B column / C column
  const int khalf = (lane >> 4) << 1;   // 0 or 2
  const float* Wp = W + col0 + r;

  v8f acc = {};
  for (int k = 0; k < K; k += 4) {
    float2 av = *(const float2*)&As[r * ldsK + k + khalf];   // ds_load_b64
    v2f a; a.x = av.x; a.y = av.y;
    v2f b;
    b.x = Wp[(size_t)(k + khalf)     * NCOLS];
    b.y = Wp[(size_t)(k + khalf + 1) * NCOLS];
    // 8 args: (neg_a, A, neg_b, B, c_mod, C, reuse_a, reuse_b)
    acc = __builtin_amdgcn_wmma_f32_16x16x4_f32(false, a, false, b,
                                                (short)0, acc, false, false);
  }

  // C/D layout: VGPR i -> row i (lanes 0-15) or row 8+i (lanes 16-31)
  const int rbase = (lane >> 4) << 3;
  float* Hp = H + (size_t)(m0 + rbase) * NCOLS + col0 + r;
#pragma unroll
  for (int i = 0; i < 8; ++i) Hp[(size_t)i * NCOLS] = acc[i];
}

// ---------------- out[i,:] = h[i,:] * dinv[i]^2  (self-loop, full overwrite) ----
__global__ void selfloop_k(const float* __restrict__ h,
                           const float* __restrict__ dinv,
                           float* __restrict__ out) {
  int node = blockIdx.x;
  float di = dinv[node];
  size_t o = (size_t)node * NCOLS + threadIdx.x;
  out[o] = h[o] * (di * di);
}

// ---------------- edge scatter: out[dst] += h[src] * dinv[src]*dinv[dst] -------
// one wave32 per edge; lane L handles columns L, L+32, ..., L+224 (coalesced)
__global__ void edge_scatter_k(const float* __restrict__ h,
                               const int* __restrict__ src,
                               const int* __restrict__ dst,
                               const float* __restrict__ dinv,
                               float* __restrict__ out, int nE) {
  int gid  = blockIdx.x * blockDim.x + threadIdx.x;
  int e    = gid >> 5;                  // uniform per wave (blockDim % 32 == 0)
  int lane = gid & 31;
  if (e >= nE) return;
  int s = src[e], d = dst[e];
  float norm = dinv[s] * dinv[d];
  const float* hs = h + (size_t)s * NCOLS;
  float*       od = out + (size_t)d * NCOLS;
#pragma unroll
  for (int j = 0; j < 8; ++j) {
    int c = lane + (j << 5);
    atomicAdd(&od[c], hs[c] * norm);    // global_atomic_add_f32, L2-resident
  }
}

// ---------------- driver ----------------
extern "C" void kernel_launch(void* const* d_in, const int* in_sizes, int n_in,
                              void* d_out, int out_size, void* d_ws, size_t ws_size,
                              hipStream_t stream) {
  const float* x  = (const float*)d_in[0];
  const int*   ei = (const int*)d_in[1];
  const float* W1 = (const float*)d_in[2];
  const float* W2 = (const float*)d_in[3];
  const float* W3 = (const float*)d_in[4];

  const int M = in_sizes[0] / 128;      // 100000 nodes (divisible by 16)
  const int E = in_sizes[1] / 2;        // 1000000 edges
  const int* src = ei;
  const int* dst = ei + E;

  float* out  = (float*)d_out;                  // doubles as aggregation buffer
  float* h    = (float*)d_ws;                   // M*256 floats (102.4 MB)
  float* dinv = h + (size_t)M * NCOLS;          // M floats

  const int T = 256;
  int nbN = (M + T - 1) / T;
  int nbE = (E + T - 1) / T;
  int nbS = (int)(((size_t)E * 32 + T - 1) / T);
  dim3 gg(M / 16, 1);

  // normalization (shared by all three layers)
  deg_init_k <<<nbN, T, 0, stream>>>(dinv, M);
  deg_count_k<<<nbE, T, 0, stream>>>(dst, dinv, E);
  dinv_k     <<<nbN, T, 0, stream>>>(dinv, M);

  // layer 1: x[M,128] @ W1 -> h ; aggregate -> out
  gemm_wmma_k   <<<gg, 512, 0, stream>>>(x, W1, h, 128, 0);
  selfloop_k    <<<M, 256, 0, stream>>>(h, dinv, out);
  edge_scatter_k<<<nbS, T, 0, stream>>>(h, src, dst, dinv, out, E);

  // layer 2: relu(out) @ W2 -> h ; aggregate -> out
  gemm_wmma_k   <<<gg, 512, 0, stream>>>(out, W2, h, 256, 1);
  selfloop_k    <<<M, 256, 0, stream>>>(h, dinv, out);
  edge_scatter_k<<<nbS, T, 0, stream>>>(h, src, dst, dinv, out, E);

  // layer 3: relu(out) @ W3 -> h ; aggregate -> out (no relu on final output)
  gemm_wmma_k   <<<gg, 512, 0, stream>>>(out, W3, h, 256, 1);
  selfloop_k    <<<M, 256, 0, stream>>>(h, dinv, out);
  edge_scatter_k<<<nbS, T, 0, stream>>>(h, src, dst, dinv, out, E);
}